// MTCNN_14542759264606
// MI455X (gfx1250) — compile-verified
//
#include <hip/hip_runtime.h>
#include <stdint.h>

#define NBOX 8192
#define NW   (NBOX / 32)        // 256 mask words per row
#define CHROWS 16               // rows per TDM chunk in the scan kernel (16 KB)
#define NMS_THR 0.7f

// -------- workspace layout (units of float unless noted) --------
//  [ 0N.. 1N) wx1   unsorted square coords
//  [ 1N.. 2N) wy1
//  [ 2N.. 3N) wx2
//  [ 3N.. 4N) wy2
//  [ 4N.. 8N) sboxes (float4: x1,y1,x2,y2) sorted by descending score
//  [ 8N.. 9N) order (int)   original index at sorted position
//  [ 9N..10N) keep_sorted (int)
//  [10N.. + NBOX*NW u32) suppression bitmask (8 MB)

typedef unsigned int v4u __attribute__((ext_vector_type(4)));
typedef int          v8i __attribute__((ext_vector_type(8)));
typedef int          v4i __attribute__((ext_vector_type(4)));

// ---------------- CDNA5 async / TDM helpers ----------------
__device__ __forceinline__ uint32_t lds_byte_off(const void* p) {
  // flat address of an LDS object: low 32 bits are the LDS byte offset
  return (uint32_t)(uintptr_t)p;
}

__device__ __forceinline__ void async_b128(uint32_t lds_addr, const void* gaddr) {
  asm volatile("global_load_async_to_lds_b128 %0, %1, off"
               :: "v"(lds_addr), "v"(gaddr) : "memory");
}
__device__ __forceinline__ void wait_async_0() {
  asm volatile("s_wait_asynccnt 0" ::: "memory");
}

// One TDM descriptor: 1-D tile of 4096 dwords (16 KB) global -> LDS.
__device__ __forceinline__ void tdm_load_chunk(const uint32_t* gbase, uint32_t lds_addr) {
  uint64_t ga = (uint64_t)(uintptr_t)gbase;
  v4u g0;
  g0.x = 1u;                                                // count=1, user mode
  g0.y = lds_addr;                                          // lds_addr (bytes)
  g0.z = (uint32_t)ga;                                      // global_addr[31:0]
  g0.w = (uint32_t)((ga >> 32) & 0x1FFFFFFu) | (2u << 30);  // global_addr[56:32] | type=2
  v8i g1;
  g1.s0 = (int)(2u << 16);        // workgroup_mask=0, data_size=2 (4 bytes)
  g1.s1 = (int)(4096u << 16);     // tensor_dim0[15:0]=4096 in word1 high half
  g1.s2 = (int)(1u << 16);        // tensor_dim0 hi=0 | tensor_dim1[15:0]=1
  g1.s3 = (int)(4096u << 16);     // tensor_dim1 hi=0 | tile_dim0=4096
  g1.s4 = 0;                      // tile_dim1=0 (1-D), tile_dim2=0
  g1.s5 = (int)4096u;             // tensor_dim0_stride low 32
  g1.s6 = 0;                      // stride hi, tensor_dim1_stride low=0
  g1.s7 = 0;
  v4i z4 = {0, 0, 0, 0};
  v8i z8 = {0, 0, 0, 0, 0, 0, 0, 0};
  __builtin_amdgcn_tensor_load_to_lds(g0, g1, z4, z4, z8, 0);
}

// ---------------- phase 0: calibrate + square ----------------
__global__ __launch_bounds__(256) void k_calib(const float* __restrict__ boxes,
                                               const float* __restrict__ offs,
                                               float* __restrict__ out,
                                               float* __restrict__ ws) {
  int i = blockIdx.x * 256 + threadIdx.x;
  if (i >= NBOX) return;
  float x1 = boxes[i * 5 + 0], y1 = boxes[i * 5 + 1];
  float x2 = boxes[i * 5 + 2], y2 = boxes[i * 5 + 3];
  float sc = boxes[i * 5 + 4];
  float w = x2 - x1 + 1.0f, h = y2 - y1 + 1.0f;
  float nx1 = x1 + w * offs[i * 4 + 0];
  float ny1 = y1 + h * offs[i * 4 + 1];
  float nx2 = x2 + w * offs[i * 4 + 2];
  float ny2 = y2 + h * offs[i * 4 + 3];
  float hh = ny2 - ny1 + 1.0f, ww = nx2 - nx1 + 1.0f;
  float ms = fmaxf(hh, ww);
  float sx1 = nx1 + 0.5f * ww - 0.5f * ms;
  float sy1 = ny1 + 0.5f * hh - 0.5f * ms;
  float sx2 = sx1 + ms - 1.0f;
  float sy2 = sy1 + ms - 1.0f;
  out[i * 5 + 0] = sx1; out[i * 5 + 1] = sy1;
  out[i * 5 + 2] = sx2; out[i * 5 + 3] = sy2;
  out[i * 5 + 4] = sc;
  ws[0 * NBOX + i] = sx1; ws[1 * NBOX + i] = sy1;
  ws[2 * NBOX + i] = sx2; ws[3 * NBOX + i] = sy2;
}

// ---------------- phase 1: stable descending argsort of scores ----------------
__global__ __launch_bounds__(1024) void k_sort(const float* __restrict__ boxes,
                                               int* __restrict__ order) {
  __shared__ uint64_t key[NBOX];     // 64 KB LDS
  int tid = threadIdx.x;
  for (int i = tid; i < NBOX; i += 1024) {
    uint32_t kb = ~__float_as_uint(boxes[i * 5 + 4]);  // scores >= 0: monotone bits
    key[i] = ((uint64_t)kb << 32) | (uint32_t)i;
  }
  __syncthreads();
  for (int size = 2; size <= NBOX; size <<= 1) {
    for (int stride = size >> 1; stride > 0; stride >>= 1) {
      for (int t = tid; t < NBOX / 2; t += 1024) {
        int i = 2 * t - (t & (stride - 1));
        int j = i + stride;
        uint64_t a = key[i], b = key[j];
        bool asc = ((i & size) == 0);
        if ((a > b) == asc) { key[i] = b; key[j] = a; }
      }
      __syncthreads();
    }
  }
  for (int i = tid; i < NBOX; i += 1024)
    order[i] = (int)(uint32_t)(key[i] & 0xFFFFFFFFu);
}

// ---------------- phase 2: gather sorted boxes as float4 AoS ----------------
__global__ __launch_bounds__(256) void k_gather(float* __restrict__ ws,
                                                const int* __restrict__ order) {
  int i = blockIdx.x * 256 + threadIdx.x;
  if (i >= NBOX) return;
  int o = order[i];
  float4 b;
  b.x = ws[0 * NBOX + o]; b.y = ws[1 * NBOX + o];
  b.z = ws[2 * NBOX + o]; b.w = ws[3 * NBOX + o];
  ((float4*)(ws + 4 * NBOX))[i] = b;
}

// ---------------- phase 3: pairwise IoU -> suppression bitmask ----------------
// grid = (NW col-blocks of 32, NBOX/256 row-blocks), block = 256 threads (1 row each)
__global__ __launch_bounds__(256) void k_mask(const float* __restrict__ ws,
                                              uint32_t* __restrict__ mask) {
  const float4* sb = (const float4*)(ws + 4 * NBOX);
  int cb = blockIdx.x;
  int r  = blockIdx.y * 256 + threadIdx.x;
  int j0 = cb * 32;
  __shared__ float4 ctile[32];                 // 512 B contiguous col tile
  int t = threadIdx.x;
  if (t < 32) {                                // wave-uniform branch: wave 0, full EXEC
    async_b128(lds_byte_off(&ctile[t]), (const void*)(sb + j0 + t));
    wait_async_0();
  }
  __syncthreads();
  float4 rb = sb[r];
  float ra = (rb.z - rb.x + 1.0f) * (rb.w - rb.y + 1.0f);
  uint32_t bits = 0u;
#pragma unroll 8
  for (int jj = 0; jj < 32; ++jj) {
    int j = j0 + jj;
    float4 cbx = ctile[jj];
    float ca = (cbx.z - cbx.x + 1.0f) * (cbx.w - cbx.y + 1.0f);
    float ix1 = fmaxf(rb.x, cbx.x), iy1 = fmaxf(rb.y, cbx.y);
    float ix2 = fminf(rb.z, cbx.z), iy2 = fminf(rb.w, cbx.w);
    float iw = fmaxf(0.0f, ix2 - ix1 + 1.0f);
    float ih = fmaxf(0.0f, iy2 - iy1 + 1.0f);
    float inter = iw * ih;
    float ov = inter / (ra + ca - inter);      // union mode
    if (j > r && ov > NMS_THR) bits |= (1u << jj);
  }
  mask[(size_t)r * NW + cb] = bits;
}

// ---------------- phase 4: sequential greedy scan (single wave32 + TDM) ----------------
__global__ __launch_bounds__(32) void k_scan(const uint32_t* __restrict__ mask,
                                             int* __restrict__ keep_sorted) {
  __shared__ uint32_t remv[NW];                // 1 KB removed-bitmap
  __shared__ uint32_t buf[2][CHROWS * NW];     // 2 x 16 KB chunk buffers
  int lane = threadIdx.x;
  for (int q = lane; q < NW; q += 32) remv[q] = 0u;
  __syncthreads();
  uint32_t lds0 = lds_byte_off(&buf[0][0]);
  uint32_t lds1 = lds_byte_off(&buf[1][0]);
  const int NCH = NBOX / CHROWS;
  tdm_load_chunk(mask, lds0);                  // prime the pipeline
  for (int c = 0; c < NCH; ++c) {
    if (c + 1 < NCH) {
      tdm_load_chunk(mask + (size_t)(c + 1) * CHROWS * NW, (c & 1) ? lds0 : lds1);
      __builtin_amdgcn_s_wait_tensorcnt(1);    // TDM in-order: chunk c is done
    } else {
      __builtin_amdgcn_s_wait_tensorcnt(0);
    }
    const uint32_t* rowbuf = buf[c & 1];
    for (int k = 0; k < CHROWS; ++k) {
      int r = c * CHROWS + k;
      uint32_t sup = (remv[r >> 5] >> (r & 31)) & 1u;   // uniform LDS broadcast
      if (!sup) {
        for (int q = lane; q < NW; q += 32)
          remv[q] |= rowbuf[k * NW + q];
      }
      if (lane == 0) keep_sorted[r] = sup ? 0 : 1;
      __syncthreads();                          // order LDS RMW across rows
    }
  }
}

// ---------------- phase 5: scatter keep mask back to original order ----------------
__global__ __launch_bounds__(256) void k_scatter(const int* __restrict__ order,
                                                 const int* __restrict__ keep_sorted,
                                                 float* __restrict__ out) {
  int i = blockIdx.x * 256 + threadIdx.x;
  if (i >= NBOX) return;
  out[5 * NBOX + order[i]] = keep_sorted[i] ? 1.0f : 0.0f;
}

extern "C" void kernel_launch(void* const* d_in, const int* in_sizes, int n_in,
                              void* d_out, int out_size, void* d_ws, size_t ws_size,
                              hipStream_t stream) {
  const float* boxes = (const float*)d_in[0];   // [N,5]
  const float* offs  = (const float*)d_in[1];   // [N,4]
  float* out = (float*)d_out;                   // [N*5] sq ++ [N] keep (float 0/1)
  float* ws  = (float*)d_ws;
  int*      order = (int*)(ws + 8 * NBOX);
  int*      keeps = (int*)(ws + 9 * NBOX);
  uint32_t* mask  = (uint32_t*)(ws + 10 * NBOX);  // 8 MB

  k_calib  <<<NBOX / 256, 256, 0, stream>>>(boxes, offs, out, ws);
  k_sort   <<<1, 1024, 0, stream>>>(boxes, order);
  k_gather <<<NBOX / 256, 256, 0, stream>>>(ws, order);
  k_mask   <<<dim3(NW, NBOX / 256), 256, 0, stream>>>(ws, mask);
  k_scan   <<<1, 32, 0, stream>>>(mask, keeps);
  k_scatter<<<NBOX / 256, 256, 0, stream>>>(order, keeps, out);
}